// PointPatchDropout_49520972923208
// MI455X (gfx1250) — compile-verified
//
#include <hip/hip_runtime.h>
#include <stdint.h>

#define BATCH 64
#define NPTS  65536
#define NC    4
#define R2    0.04f   // PATCH_RADIUS^2

typedef __attribute__((ext_vector_type(2))) float v2f;
typedef __attribute__((ext_vector_type(8))) float v8f;

// ds_swizzle group-of-32 XOR lane swap: offset = (xor_mask<<10) | and_mask 0x1F
__device__ __forceinline__ float swz_xor1(float x) {
  return __int_as_float(__builtin_amdgcn_ds_swizzle(__float_as_int(x), 0x041F));
}
__device__ __forceinline__ float swz_xor2(float x) {
  return __int_as_float(__builtin_amdgcn_ds_swizzle(__float_as_int(x), 0x081F));
}

// ---------------------------------------------------------------------------
// Kernel 1: Furthest point sampling (exact reference formula, exact argmax
// tie-break). One 1024-thread block per batch; running min-distance kept in
// registers (64 points/thread); block argmax via packed u64 keys in LDS.
// Stores per batch: centers[b*16 + s*4 + {x,y,z,|c|^2}].
// ---------------------------------------------------------------------------
__global__ __launch_bounds__(1024)
void fps_kernel(const float* __restrict__ pts, float* __restrict__ centers) {
  const int b = blockIdx.x;
  const int t = threadIdx.x;
  __shared__ unsigned long long red[1024];
  __shared__ int curIdx;

  float md[64];
#pragma unroll
  for (int i = 0; i < 64; ++i) md[i] = 1e10f;
  if (t == 0) curIdx = 0;
  __syncthreads();

  const float* base = pts + (size_t)b * NPTS * 6;

  for (int s = 0; s < NC; ++s) {
    const int ci = curIdx;
    const float cx = base[(size_t)ci * 6 + 0];
    const float cy = base[(size_t)ci * 6 + 1];
    const float cz = base[(size_t)ci * 6 + 2];
    if (t == 0) {
      centers[b * 16 + s * 4 + 0] = cx;
      centers[b * 16 + s * 4 + 1] = cy;
      centers[b * 16 + s * 4 + 2] = cz;
      centers[b * 16 + s * 4 + 3] = cx * cx + cy * cy + cz * cz;
    }
    unsigned long long best = 0ull;
#pragma unroll
    for (int i = 0; i < 64; ++i) {
      const int n = t + i * 1024;
      const float dx = base[(size_t)n * 6 + 0] - cx;
      const float dy = base[(size_t)n * 6 + 1] - cy;
      const float dz = base[(size_t)n * 6 + 2] - cz;
      const float d = dx * dx + dy * dy + dz * dz;
      md[i] = fminf(md[i], d);
      // non-negative floats: bit pattern is order-preserving.
      // ties -> larger (0xFFFFFFFF - n) wins -> smallest n (jnp.argmax rule).
      const unsigned long long key =
          ((unsigned long long)__float_as_uint(md[i]) << 32) |
          (unsigned long long)(0xFFFFFFFFu - (unsigned)n);
      best = (key > best) ? key : best;
    }
    red[t] = best;
    __syncthreads();
    for (int off = 512; off > 0; off >>= 1) {
      if (t < off) {
        const unsigned long long o = red[t + off];
        if (o > red[t]) red[t] = o;
      }
      __syncthreads();
    }
    if (t == 0) curIdx = (int)(0xFFFFFFFFu - (unsigned)(red[0] & 0xFFFFFFFFull));
    __syncthreads();
  }
}

// ---------------------------------------------------------------------------
// Kernel 2: keep mask via WMMA. D = A(16x4) * B(4x16) + 0 with
//   A row m = [x_m, y_m, z_m, 1],  B col n = [-2cx, -2cy, -2cz, |c|^2]
// so D[m][n] = -2 x.c + |c|^2 and dist^2 = |x|^2 + D. Only columns 0..3 are
// real centers; columns 4..15 are zeros and never enter the reduction.
// A layout (f32 16x4): lanes 0-15 hold (K0,K1)=(x,y) of point m=lane,
//                      lanes 16-31 hold (K2,K3)=(z,1) of point m=lane-16.
// B layout (f32 4x16): lane j = column n; half selects K pair like A.
// D layout: VGPR r, half h -> point m = r + 8h, lane j -> column n.
// Min over the 4 real columns: ds_swizzle XOR {1,2} folds lanes j in {0..3};
// finalize lanes (j==0 in each half) then hold min over centers 0..3.
// ---------------------------------------------------------------------------
__global__ __launch_bounds__(256)
void keep_kernel(const float* __restrict__ pts,
                 const float* __restrict__ centers,
                 uint8_t* __restrict__ keep) {
  const int lane  = threadIdx.x & 31;
  const int gw    = blockIdx.x * 8 + (threadIdx.x >> 5);  // global wave id
  const int b     = gw >> 12;                             // 4096 waves / batch
  const int pBase = (gw & 4095) << 4;                     // 16 points / wave
  const int h     = lane >> 4;
  const int j     = lane & 15;

  // A operand
  const float* prow = pts + ((size_t)b * NPTS + (size_t)(pBase + j)) * 6;
  v2f a;
  if (h == 0) { a.x = prow[0]; a.y = prow[1]; }
  else        { a.x = prow[2]; a.y = 1.0f;    }

  // B operand (columns >= NC are zero; harmless, excluded from the min)
  v2f bm = {0.0f, 0.0f};
  if (j < NC) {
    const float* c = centers + b * 16 + j * 4;
    if (h == 0) { bm.x = -2.0f * c[0]; bm.y = -2.0f * c[1]; }
    else        { bm.x = -2.0f * c[2]; bm.y = c[3];         }
  }

  v8f acc = {};
  acc = __builtin_amdgcn_wmma_f32_16x16x4_f32(
      /*neg_a=*/false, a, /*neg_b=*/false, bm,
      /*c_mod=*/(short)0, acc, /*reuse_a=*/false, /*reuse_b=*/false);

  // min over centers 0..3: fold lanes within each XOR-4 group
  float mn[8];
#pragma unroll
  for (int r = 0; r < 8; ++r) {
    float v = acc[r];
    v = fminf(v, swz_xor1(v));
    v = fminf(v, swz_xor2(v));
    mn[r] = v;   // valid in lanes with (j & 3) arbitrary, we read j==0
  }

  // finalize: one lane per half adds |x|^2 and writes 8 keep flags
  if (j == 0) {
#pragma unroll
    for (int r = 0; r < 8; ++r) {
      const int p = pBase + r + 8 * h;
      const float* q = pts + ((size_t)b * NPTS + p) * 6;
      const float xx = q[0] * q[0] + q[1] * q[1] + q[2] * q[2];
      keep[(size_t)b * NPTS + p] = ((mn[r] + xx) >= R2) ? 1 : 0;
    }
  }
}

// ---------------------------------------------------------------------------
// Kernel 3: stable compaction of kept indices. One block per batch; 1024-wide
// Hillis-Steele inclusive scan per chunk, running base offset across chunks.
// ---------------------------------------------------------------------------
__global__ __launch_bounds__(1024)
void compact_kernel(const uint8_t* __restrict__ keep, int* __restrict__ cmp,
                    int* __restrict__ nvalid) {
  const int b = blockIdx.x;
  const int t = threadIdx.x;
  __shared__ int sc[1024];
  const uint8_t* kp = keep + (size_t)b * NPTS;
  int* cp = cmp + (size_t)b * NPTS;
  int basecnt = 0;
  for (int c0 = 0; c0 < NPTS; c0 += 1024) {
    const int n = c0 + t;
    const int k = kp[n];
    sc[t] = k;
    __syncthreads();
    for (int off = 1; off < 1024; off <<= 1) {
      const int v = (t >= off) ? sc[t - off] : 0;
      __syncthreads();
      sc[t] += v;
      __syncthreads();
    }
    if (k) cp[basecnt + sc[t] - 1] = n;
    basecnt += sc[1023];
    __syncthreads();
  }
  if (t == 0) nvalid[b] = basecnt;
}

// ---------------------------------------------------------------------------
// Kernel 4: cyclic gather of kept rows into the output (24-byte rows moved as
// three 8-byte float2 chunks; rows are 8-byte aligned).
// ---------------------------------------------------------------------------
__global__ __launch_bounds__(256)
void gather_kernel(const float* __restrict__ pts, const int* __restrict__ cmp,
                   const int* __restrict__ nvalid, float* __restrict__ out) {
  const int g  = blockIdx.x * 256 + threadIdx.x;  // [0, BATCH*NPTS)
  const int b  = g >> 16;
  const int jj = g & (NPTS - 1);
  const int nv = nvalid[b];
  float2* o = (float2*)(out + ((size_t)b * NPTS + jj) * 6);
  if (nv <= 0) {
    const float2 z = make_float2(0.f, 0.f);
    o[0] = z; o[1] = z; o[2] = z;
    return;
  }
  const int src = cmp[(size_t)b * NPTS + (jj % nv)];
  const float2* s = (const float2*)(pts + ((size_t)b * NPTS + src) * 6);
  o[0] = s[0]; o[1] = s[1]; o[2] = s[2];
}

// ---------------------------------------------------------------------------
extern "C" void kernel_launch(void* const* d_in, const int* in_sizes, int n_in,
                              void* d_out, int out_size, void* d_ws, size_t ws_size,
                              hipStream_t stream) {
  const float* pts = (const float*)d_in[0];
  float* out = (float*)d_out;

  // workspace layout (~20 MB):
  //   [0, B*N)                 keep flags (u8)
  //   [B*N, B*N*5)             compact indices (i32)
  //   [B*N*5, +B*16*4)         centers (f32: x,y,z,|c|^2 per center)
  //   then                     num_valid (i32 per batch)
  char* ws = (char*)d_ws;
  uint8_t* keep   = (uint8_t*)ws;
  int*     cmp    = (int*)(ws + (size_t)BATCH * NPTS);
  float*   ctrs   = (float*)(ws + (size_t)BATCH * NPTS * 5);
  int*     nvalid = (int*)(ws + (size_t)BATCH * NPTS * 5 + (size_t)BATCH * 16 * 4);

  fps_kernel<<<BATCH, 1024, 0, stream>>>(pts, ctrs);
  keep_kernel<<<(BATCH * NPTS) / 128, 256, 0, stream>>>(pts, ctrs, keep);
  compact_kernel<<<BATCH, 1024, 0, stream>>>(keep, cmp, nvalid);
  gather_kernel<<<(BATCH * NPTS) / 256, 256, 0, stream>>>(pts, cmp, nvalid, out);
}